// RLayerNorm_89258010345528
// MI455X (gfx1250) — compile-verified
//
#include <hip/hip_runtime.h>
#include <hip/hip_bf16.h>
#include <stdint.h>

#define HDIM   128
#define WDIM   128
#define NPIX   (HDIM * WDIM)
#define STRIDE 132              // 128 data floats + 4 pad floats per row (TDM pad)
#define BLK    256
#define EPS    1e-5f

typedef unsigned int v4u __attribute__((ext_vector_type(4)));
typedef int          v8i __attribute__((ext_vector_type(8)));
typedef int          v4i __attribute__((ext_vector_type(4)));
typedef float        v4f __attribute__((ext_vector_type(4)));

__global__ __launch_bounds__(BLK) void rlayernorm_kernel(
    const float* __restrict__ x, const float* __restrict__ weight,
    const float* __restrict__ bias, const float* __restrict__ lambda_p,
    float* __restrict__ out, int C)
{
    __shared__ __align__(16) float tile[HDIM * STRIDE];   // ~66 KB plane stage
    __shared__ float red[8][3];
    __shared__ float bc[2];

    const int  tid   = threadIdx.x;
    const int  plane = blockIdx.x;
    const float* xp  = x + (size_t)plane * NPIX;

    // ---- Stage plane into LDS ------------------------------------------------
#if __has_builtin(__builtin_amdgcn_tensor_load_to_lds)
    // TDM: 2D tile 128x128 f32, with LDS padding of 4 DWORDs every 128 DWORDs
    // so LDS row stride becomes 132 floats (bank-conflict-free stencil reads).
    if (tid < 32) {  // one wave issues the descriptor (EXEC ignored by TDM)
        uint64_t g      = (uint64_t)(uintptr_t)xp;
        uint32_t ldsoff = (uint32_t)(uintptr_t)&tile[0];  // flat LDS addr low 32b
        v4u g0;
        g0[0] = 1u;                                        // count=1, no gather
        g0[1] = ldsoff;                                    // lds_addr
        g0[2] = (uint32_t)g;                               // global_addr[31:0]
        g0[3] = ((uint32_t)(g >> 32) & 0x01FFFFFFu) | (2u << 30); // addr hi | type=2
        v8i g1;
        // data_size=2 (4B) | pad_enable | pad_interval=6 (128 dw) | pad_amount=3 (4 dw)
        g1[0] = (int)((2u << 16) | (1u << 20) | (6u << 22) | (3u << 25));
        g1[1] = (int)(128u << 16);          // tensor_dim0 = 128 (lo16)
        g1[2] = (int)(128u << 16);          // tensor_dim1 = 128 (lo16)
        g1[3] = (int)(128u << 16);          // tile_dim0   = 128
        g1[4] = 128;                        // tile_dim1   = 128, tile_dim2 = 0
        g1[5] = 128;                        // tensor_dim0_stride = 128
        g1[6] = (int)(16384u << 16);        // tensor_dim1_stride = 16384 (lo16)
        g1[7] = 0;
        v4i g2 = {0, 0, 0, 0};
        v4i g3 = {0, 0, 0, 0};
        // cpol = 1: TH=NT (stream-once input; its only reuse lives in LDS)
#if defined(__clang_major__) && (__clang_major__ >= 23)
        v8i g4 = {0, 0, 0, 0, 0, 0, 0, 0};
        __builtin_amdgcn_tensor_load_to_lds(g0, g1, g2, g3, g4, 1);
#else
        __builtin_amdgcn_tensor_load_to_lds(g0, g1, g2, g3, 1);
#endif
        __builtin_amdgcn_s_wait_tensorcnt(0);
    }
#else
    // Fallback: cooperative vectorized load with manual row padding
    for (int k = tid; k < NPIX / 4; k += BLK) {
        int row = k >> 5, col = (k & 31) << 2;
        *reinterpret_cast<v4f*>(&tile[row * STRIDE + col]) =
            *reinterpret_cast<const v4f*>(xp + (size_t)k * 4);
    }
#endif
    __syncthreads();

    // ---- Fused stats pass ----------------------------------------------------
    // Thread t owns row (t&127), column half (t>>7). Sliding 3-tap column sums
    // give the 3x3 window in 3 LDS reads/pixel. Reflect: -1 -> 1, 128 -> 126.
    const int row  = tid & 127;
    const int half = tid >> 7;
    const int c0   = half * 64;
    const int rm   = (row == 0)   ? 1   : row - 1;
    const int rp   = (row == 127) ? 126 : row + 1;
    const float* Lm = &tile[rm  * STRIDE];
    const float* L0 = &tile[row * STRIDE];
    const float* Lp = &tile[rp  * STRIDE];

    float p1, p2, q1, q2, qc;
    {   // warm-up columns: reflect(c0-1), c0
        int cm = c0 - 1; if (cm < 0) cm = 1;
        float a = Lm[cm], b = L0[cm], d = Lp[cm];
        p1 = a + b + d;
        p2 = fmaf(a, a, fmaf(b, b, d * d));
        a = Lm[c0]; b = L0[c0]; d = Lp[c0];
        q1 = a + b + d;
        q2 = fmaf(a, a, fmaf(b, b, d * d));
        qc = b;
    }

    float sX = 0.0f, sX2 = 0.0f, sLV = 0.0f;
    const float inv9 = 1.0f / 9.0f;
    #pragma unroll 4
    for (int c = c0; c < c0 + 64; ++c) {
        int cn = c + 1; if (cn > 127) cn = 126;           // reflect right edge
        float a = Lm[cn], b = L0[cn], d = Lp[cn];
        float n1 = a + b + d;
        float n2 = fmaf(a, a, fmaf(b, b, d * d));
        float s1 = p1 + q1 + n1;
        float s2 = p2 + q2 + n2;
        float m  = s1 * inv9;
        float lv = fmaf(-m, m, s2 * inv9);                // E[x^2]_loc - m^2
        sLV += fmaxf(lv, 0.0f);
        sX  += qc;
        sX2  = fmaf(qc, qc, sX2);
        p1 = q1; p2 = q2; q1 = n1; q2 = n2; qc = b;
    }

    // ---- Block reduction (wave32 shuffles, then cross-wave via LDS) ----------
    #pragma unroll
    for (int o = 16; o > 0; o >>= 1) {
        sX  += __shfl_down(sX,  o, 32);
        sX2 += __shfl_down(sX2, o, 32);
        sLV += __shfl_down(sLV, o, 32);
    }
    const int wave = tid >> 5, lane = tid & 31;
    if (lane == 0) { red[wave][0] = sX; red[wave][1] = sX2; red[wave][2] = sLV; }
    __syncthreads();

    if (tid == 0) {
        float a = 0.0f, b2 = 0.0f, c3 = 0.0f;
        #pragma unroll
        for (int i = 0; i < 8; ++i) { a += red[i][0]; b2 += red[i][1]; c3 += red[i][2]; }
        const float invN = 1.0f / (float)NPIX;
        float mean  = a * invN;
        float var   = fmaf(-mean, mean, b2 * invN);
        float noise = c3 * invN;
        float lam   = fminf(fmaxf(lambda_p[0], 0.001f), 1.0f);
        float stdv  = sqrtf(var + EPS);
        float nsc   = 1.0f + lam * noise / (var + EPS);
        float inv   = 1.0f / (stdv * nsc + EPS);
        int   ch    = plane % C;                 // plane = b*C + c
        float wch   = weight[ch];
        float bch   = bias[ch];
        float alpha = wch * inv;
        bc[0] = alpha;
        bc[1] = fmaf(-mean, alpha, bch);         // out = x*alpha + beta
    }
    __syncthreads();
    const float alpha = bc[0], beta = bc[1];

    // ---- Normalize + coalesced non-temporal float4 store ---------------------
    // Output is write-once (256 MB, exceeds 192 MB L2): NT keeps L2 for streams.
    v4f* outp = reinterpret_cast<v4f*>(out + (size_t)plane * NPIX);
    #pragma unroll 4
    for (int k = 0; k < 16; ++k) {
        int i4  = k * BLK + tid;                 // float4 index 0..4095
        int r   = i4 >> 5;                       // 32 float4 per row
        int col = (i4 & 31) << 2;                // 16B-aligned (row*132*4 % 16 == 0)
        v4f v = *reinterpret_cast<const v4f*>(&tile[r * STRIDE + col]);
        v4f o;
        o.x = fmaf(v.x, alpha, beta);
        o.y = fmaf(v.y, alpha, beta);
        o.z = fmaf(v.z, alpha, beta);
        o.w = fmaf(v.w, alpha, beta);
        __builtin_nontemporal_store(o, &outp[i4]);
    }
}

extern "C" void kernel_launch(void* const* d_in, const int* in_sizes, int n_in,
                              void* d_out, int out_size, void* d_ws, size_t ws_size,
                              hipStream_t stream) {
    const float* x      = (const float*)d_in[0];
    const float* weight = (const float*)d_in[1];
    const float* bias   = (const float*)d_in[2];
    const float* lam    = (const float*)d_in[3];
    float*       out    = (float*)d_out;
    (void)d_ws; (void)ws_size; (void)n_in; (void)out_size;

    const int planes = in_sizes[0] / NPIX;       // B*C = 4096
    const int C      = in_sizes[1];              // 128
    rlayernorm_kernel<<<planes, BLK, 0, stream>>>(x, weight, bias, lam, out, C);
}